// ForwardTacotron_33285996544563
// MI455X (gfx1250) — compile-verified
//
#include <hip/hip_runtime.h>

// Length regulator: y[b,t,:] = x[b, idx[b,t], :], idx = searchsorted(cumsum(dur), t, right) clipped.
// B=32, T=256, D=512, MAX_LEN=2048. Pure memory-bound gather: ~151 MB traffic, ~6.5us floor @23.3TB/s.
// Data path: gfx1250 async LDS engine (GLOBAL_LOAD_ASYNC_TO_LDS_B128 / GLOBAL_STORE_ASYNC_FROM_LDS_B128,
// ASYNCcnt-tracked) — bulk copy without a VGPR round-trip.

#define LR_B      32
#define LR_T      256
#define LR_D      512
#define LR_MAXLEN 2048
#define LR_VEC    (LR_D / 4)            // 128 float4 per feature row
#define ROWS_PB   8                     // output rows per block
#define F4_PB     (ROWS_PB * LR_VEC)    // 1024 float4 per block
#define THREADS   256
#define F4_PER_THREAD (F4_PB / THREADS) // 4

typedef float f4  __attribute__((ext_vector_type(4)));
typedef int   v4i __attribute__((ext_vector_type(4)));

#if defined(__has_builtin)
#if __has_builtin(__builtin_amdgcn_global_load_async_to_lds_b128) && \
    __has_builtin(__builtin_amdgcn_global_store_async_from_lds_b128)
#define LR_ASYNC 1
#endif
#endif
#ifndef LR_ASYNC
#define LR_ASYNC 0
#endif

#if LR_ASYNC
typedef __attribute__((address_space(1))) v4i* g128_t;  // global b128 pointer
typedef __attribute__((address_space(3))) v4i* l128_t;  // LDS b128 pointer

__device__ __forceinline__ void lr_wait_async_zero() {
#if __has_builtin(__builtin_amdgcn_s_wait_asynccnt)
  __builtin_amdgcn_s_wait_asynccnt(0);
#else
  asm volatile("s_wait_asynccnt 0" ::: "memory");
#endif
}
#endif

// ---------------- Kernel 1: per-batch inclusive cumsum of durations ----------------
__global__ __launch_bounds__(LR_T) void lr_cumsum_kernel(const int* __restrict__ dur,
                                                         int* __restrict__ cum) {
  __shared__ int s[LR_T];
  const int t = threadIdx.x;
  const int b = blockIdx.x;
  int d = dur[b * LR_T + t];
  if (d < 0) d = 0;
  s[t] = d;
  __syncthreads();
#pragma unroll
  for (int off = 1; off < LR_T; off <<= 1) {
    int v = (t >= off) ? s[t - off] : 0;
    __syncthreads();
    s[t] += v;
    __syncthreads();
  }
  cum[b * LR_T + t] = s[t];
}

// ---------------- Kernel 2: gather 8 output rows per block via async LDS copy ----------------
__global__ __launch_bounds__(THREADS) void lr_gather_kernel(const float* __restrict__ x,
                                                            const int* __restrict__ cum,
                                                            float* __restrict__ out) {
  __shared__ int scum[LR_T];
  __shared__ int ridx[ROWS_PB];
#if LR_ASYNC
  __shared__ f4 buf[F4_PB];  // 16 KB staging for async global->LDS->global
#endif
  const int tid  = threadIdx.x;
  const int row0 = blockIdx.x * ROWS_PB;   // first global output row (aligned: all 8 share batch)
  const int b    = row0 >> 11;             // row0 / LR_MAXLEN

  scum[tid] = cum[b * LR_T + tid];
  __syncthreads();

  if (tid < ROWS_PB) {
    const int f = (row0 & (LR_MAXLEN - 1)) + tid;  // frame within batch
    int lo = 0, hi = LR_T;
#pragma unroll
    for (int step = 0; step < 8; ++step) {         // 256-entry range -> exactly 8 halvings
      const int mid = (lo + hi) >> 1;
      if (scum[mid] <= f) lo = mid + 1; else hi = mid;
    }
    ridx[tid] = (lo > LR_T - 1) ? (LR_T - 1) : lo;
  }
  __syncthreads();

  const f4* __restrict__ x4 = (const f4*)x;
  f4* __restrict__ o4 = (f4*)out;

#if LR_ASYNC
  // Issue all B128 async loads (global -> LDS), one 16B slot per (thread, k).
#pragma unroll
  for (int k = 0; k < F4_PER_THREAD; ++k) {
    const int slot = tid + k * THREADS;            // 0..1023
    const int r    = slot >> 7;                    // local row 0..7
    const int lane = slot & (LR_VEC - 1);          // float4 within row
    const int i    = ridx[r];
    const f4* src  = x4 + ((b * LR_T + i) * LR_VEC + lane);
    __builtin_amdgcn_global_load_async_to_lds_b128((g128_t)(void*)src, (l128_t)&buf[slot], 0, 0);
  }
  lr_wait_async_zero();  // per-wave: our loads have landed in our LDS slots
  // Stream B128 async stores (LDS -> global); s_endpgm's implicit wait-idle drains them.
#pragma unroll
  for (int k = 0; k < F4_PER_THREAD; ++k) {
    const int slot = tid + k * THREADS;
    f4* dst = o4 + (row0 * LR_VEC + slot);
    __builtin_amdgcn_global_store_async_from_lds_b128((g128_t)(void*)dst, (l128_t)&buf[slot], 0, 0);
  }
#else
  // Fallback: direct VGPR path, nontemporal stores keep the 134MB write stream out of L2.
#pragma unroll
  for (int k = 0; k < F4_PER_THREAD; ++k) {
    const int slot = tid + k * THREADS;
    const int r    = slot >> 7;
    const int lane = slot & (LR_VEC - 1);
    const int i    = ridx[r];
    f4 v = x4[(b * LR_T + i) * LR_VEC + lane];
    __builtin_nontemporal_store(v, o4 + (row0 * LR_VEC + slot));
  }
#endif
}

extern "C" void kernel_launch(void* const* d_in, const int* in_sizes, int n_in,
                              void* d_out, int out_size, void* d_ws, size_t ws_size,
                              hipStream_t stream) {
  (void)in_sizes; (void)n_in; (void)out_size; (void)ws_size;
  const float* x  = (const float*)d_in[0];   // [B, T, D] fp32
  const int*  dur = (const int*)d_in[1];     // [B, T] int32
  float* out = (float*)d_out;                // [B, MAX_LEN, D] fp32
  int* cum = (int*)d_ws;                     // B*T ints = 32 KB scratch

  lr_cumsum_kernel<<<LR_B, LR_T, 0, stream>>>(dur, cum);
  lr_gather_kernel<<<(LR_B * LR_MAXLEN) / ROWS_PB, THREADS, 0, stream>>>(x, cum, out);
}